// ShiftInvariantLoss_2645699854800
// MI455X (gfx1250) — compile-verified
//
#include <hip/hip_runtime.h>
#include <hip/hip_bf16.h>

typedef __attribute__((ext_vector_type(2))) float v2f;
typedef __attribute__((ext_vector_type(8))) float v8f;

#define BATCH 32
#define N1    128          // input H/W
#define NT    140          // target H/W
#define S1    13           // valid shifts per dim
#define NPIX  (N1 * N1)    // 16384
#define TPIX  (NT * NT)    // 19600

// ---------------------------------------------------------------------------
// Workspace layout (floats):
//   [0,144)        zero row (A-fragment pointer target for padded rows)
//   [144,176)      A_b softplus sums
//   [176,176+8192) corr tiles (16x16 per batch)
//   [8368,8400)    per-batch min loss
// ---------------------------------------------------------------------------
#define WS_Z     0
#define WS_A     144
#define WS_CORR  176
#define WS_LOSS  (WS_CORR + BATCH * 256)

// K1: A_b = sum over pixels of max(x,0)+log1p(exp(-|x|)). One block per batch.
//     Block 0 additionally zeroes the shared zero-row used by K2.
__global__ void sil_softplus_sum(const float* __restrict__ x, float* __restrict__ A,
                                 float* __restrict__ zrow) {
    const int b = blockIdx.x, tid = threadIdx.x;
    if (b == 0 && tid < 144) zrow[tid] = 0.0f;
    const float* X = x + b * NPIX;
    float s = 0.0f;
    for (int e = tid; e < NPIX; e += 256) {
        float v = X[e];
        s += fmaxf(v, 0.0f) + log1pf(expf(-fabsf(v)));
    }
    __shared__ float r[256];
    r[tid] = s;
    __syncthreads();
    for (int st = 128; st > 0; st >>= 1) {
        if (tid < st) r[tid] += r[tid + st];
        __syncthreads();
    }
    if (tid == 0) A[b] = r[0];
}

// K2: 2-D cross-correlation via V_WMMA_F32_16X16X4_F32.
// corr[i][j] = sum_p sum_K t[p+i][K] * xpad[p][K-j]
// One block (8 waves) per batch; wave w handles p = w, w+8, ...
// Boundary handling is hoisted out of the K loop:
//   - A rows beyond t point at a global zero row (pointer select per p)
//   - B reads a per-wave LDS row of x padded with 16 zeros on each side
__global__ void sil_corr_wmma(const float* __restrict__ x, const float* __restrict__ t,
                              const float* __restrict__ zrow, float* __restrict__ corr) {
    const int b    = blockIdx.x;
    const int lane = threadIdx.x & 31;
    const int wave = threadIdx.x >> 5;     // 0..7
    const int m    = lane & 15;            // A: row i; B: col j
    const int half = lane >> 4;            // K striping (0 or 1)

    const float* X = x + b * NPIX;         // [128][128]
    const float* T = t + b * TPIX;         // [140][140]

    __shared__ float xbuf[8][160];         // per-wave padded x row
    // zero the +/-16 element pads once (static for whole p loop)
    if (lane < 16) {
        xbuf[wave][lane]       = 0.0f;
        xbuf[wave][144 + lane] = 0.0f;
    }

    const int xb = 16 + 2 * half - m;      // LDS base index for this lane's B pair

    v8f acc = {};
    for (int p = wave; p < N1; p += 8) {
        // stage x row p into LDS interior: one float4 per lane
        const float4 xv = *(const float4*)(X + p * N1 + 4 * lane);
        *(float4*)&xbuf[wave][16 + 4 * lane] = xv;

        // A-fragment row pointer: valid t row or the zero row (select per p)
        const int arow = p + m;
        const float* ta = ((arow < NT) ? (T + arow * NT) : zrow) + 2 * half;
        const float* xr = &xbuf[wave][xb];

        #pragma unroll
        for (int k0 = 0; k0 < NT; k0 += 4) {    // K = 140, 35 WMMAs
            v2f av; av.x = ta[k0]; av.y = ta[k0 + 1];     // global b64
            v2f bv; bv.x = xr[k0]; bv.y = xr[k0 + 1];     // ds 2addr
            acc = __builtin_amdgcn_wmma_f32_16x16x4_f32(
                false, av, false, bv, (short)0, acc, false, false);
        }
    }

    // Reduce 8 per-wave partial C tiles through LDS (fixed order -> deterministic)
    __shared__ float red[8 * 256];
    #pragma unroll
    for (int v = 0; v < 8; ++v) red[wave * 256 + v * 32 + lane] = acc[v];
    __syncthreads();

    const int tid = threadIdx.x;
    float s = 0.0f;
    #pragma unroll
    for (int w = 0; w < 8; ++w) s += red[w * 256 + tid];
    // C/D layout: VGPR v, lanes 0-15 -> M=v, lanes 16-31 -> M=v+8; N = lane&15
    const int v  = tid >> 5;
    const int l  = tid & 31;
    const int Mi = v + 8 * (l >> 4);
    const int Nj = l & 15;
    corr[b * 256 + Mi * 16 + Nj] = s;
}

// K3: per-batch argmax over valid 13x13 shifts, min-loss, and window copy.
__global__ void sil_finalize(const float* __restrict__ t, const float* __restrict__ corr,
                             const float* __restrict__ A, float* __restrict__ loss,
                             float* __restrict__ out_labels) {
    const int b = blockIdx.x, tid = threadIdx.x;
    __shared__ float v[256];
    __shared__ int   ix[256];
    float best = -3.4e38f;
    int   bidx = 0;
    if (tid < S1 * S1) {
        const int i = tid / S1, j = tid % S1;
        best = corr[b * 256 + i * 16 + j];
        bidx = tid;
    }
    v[tid] = best; ix[tid] = bidx;
    __syncthreads();
    for (int st = 128; st > 0; st >>= 1) {
        if (tid < st) {
            const bool take = (v[tid + st] > v[tid]) ||
                              (v[tid + st] == v[tid] && ix[tid + st] < ix[tid]);
            if (take) { v[tid] = v[tid + st]; ix[tid] = ix[tid + st]; }
        }
        __syncthreads();
    }
    __shared__ int sbi, sbj;
    if (tid == 0) {
        sbi = ix[0] / S1;
        sbj = ix[0] % S1;
        loss[b] = (A[b] - v[0]) * (1.0f / (float)NPIX);
    }
    __syncthreads();
    const float* T = t + b * TPIX;
    const int bi = sbi, bj = sbj;
    for (int e = tid; e < NPIX; e += 256) {
        const int r = e >> 7, c = e & (N1 - 1);
        out_labels[b * NPIX + e] = T[(r + bi) * NT + (c + bj)];
    }
}

// K4: fixed-order sum of 32 per-batch losses -> out[0] (deterministic, no atomics)
__global__ void sil_total(const float* __restrict__ loss, float* __restrict__ out0) {
    if (threadIdx.x == 0 && blockIdx.x == 0) {
        float s = 0.0f;
        for (int b = 0; b < BATCH; ++b) s += loss[b];
        out0[0] = s;
    }
}

extern "C" void kernel_launch(void* const* d_in, const int* in_sizes, int n_in,
                              void* d_out, int out_size, void* d_ws, size_t ws_size,
                              hipStream_t stream) {
    const float* x = (const float*)d_in[0];   // [32,1,128,128]
    const float* t = (const float*)d_in[1];   // [32,1,140,140]
    float* out = (float*)d_out;               // out[0]=total_loss, out[1..] = labels
    float* ws  = (float*)d_ws;

    float* wsZ    = ws + WS_Z;
    float* wsA    = ws + WS_A;
    float* wsCorr = ws + WS_CORR;
    float* wsLoss = ws + WS_LOSS;

    sil_softplus_sum<<<BATCH, 256, 0, stream>>>(x, wsA, wsZ);
    sil_corr_wmma  <<<BATCH, 256, 0, stream>>>(x, t, wsZ, wsCorr);
    sil_finalize   <<<BATCH, 256, 0, stream>>>(t, wsCorr, wsA, wsLoss, out + 1);
    sil_total      <<<1, 32, 0, stream>>>(wsLoss, out);
}